// ScaledDotProductAttention_5634997092986
// MI455X (gfx1250) — compile-verified
//
#include <hip/hip_runtime.h>
#include <stdint.h>

typedef __attribute__((ext_vector_type(2))) float v2f;
typedef __attribute__((ext_vector_type(8))) float v8f;

#define B_   4
#define H_   4
#define L1_  16
#define L2_  512
#define DK_  32
#define QT_  (L2_ / 16)                 // 32 query tiles per problem
#define FT_  4                          // key tiles fused per softmax update
#define NKG_ (L2_ / (16 * FT_))         // 8 outer key groups

#define LOG2E_   1.4426950408889634f
#define SCALE2_  (0.17677669529663687f * LOG2E_)   // (1/sqrt(32)) * log2(e)
#define NEGINF2_ (-1.0e9f * LOG2E_)                // masked logit in log2 domain

// ---- DPP16 helpers: reduction permutations within each 16-lane half ----
__device__ __forceinline__ float dpp_xor1(float x) {   // quad_perm [1,0,3,2]
    int i = __builtin_bit_cast(int, x);
    return __builtin_bit_cast(float, __builtin_amdgcn_update_dpp(i, i, 0xB1, 0xF, 0xF, true));
}
__device__ __forceinline__ float dpp_xor2(float x) {   // quad_perm [2,3,0,1]
    int i = __builtin_bit_cast(int, x);
    return __builtin_bit_cast(float, __builtin_amdgcn_update_dpp(i, i, 0x4E, 0xF, 0xF, true));
}
__device__ __forceinline__ float dpp_hmirror(float x) { // row_half_mirror (lane^7 in 8)
    int i = __builtin_bit_cast(int, x);
    return __builtin_bit_cast(float, __builtin_amdgcn_update_dpp(i, i, 0x141, 0xF, 0xF, true));
}
__device__ __forceinline__ float dpp_mirror(float x) {  // row_mirror (lane^15 in 16)
    int i = __builtin_bit_cast(int, x);
    return __builtin_bit_cast(float, __builtin_amdgcn_update_dpp(i, i, 0x140, 0xF, 0xF, true));
}
__device__ __forceinline__ float rowmax16(float v) {
    v = fmaxf(v, dpp_xor1(v));
    v = fmaxf(v, dpp_xor2(v));
    v = fmaxf(v, dpp_hmirror(v));
    v = fmaxf(v, dpp_mirror(v));
    return v;
}
__device__ __forceinline__ float rowsum16(float v) {
    v += dpp_xor1(v);
    v += dpp_xor2(v);
    v += dpp_hmirror(v);
    v += dpp_mirror(v);
    return v;
}

// One wave32 per (problem, 16-query tile). Flash-attention single pass,
// 64 keys per softmax update, both GEMMs on V_WMMA_F32_16X16X4_F32.
__global__ __launch_bounds__(32)
void sdpa_f32_wmma_kernel(const float* __restrict__ Q,
                          const float* __restrict__ K,
                          const float* __restrict__ V,
                          const uint8_t* __restrict__ M,
                          float* __restrict__ O)
{
    // Per-sub-tile P staging buffers; 18-float rows keep v2f reads 8B-aligned
    __shared__ __align__(16) float Pl[FT_][16][18];

    const int lane = threadIdx.x;       // 0..31
    const int g    = lane >> 4;         // half-wave group (0/1)
    const int ln   = lane & 15;

    const int qt   = blockIdx.x & (QT_ - 1);   // query tile
    const int prob = blockIdx.x >> 5;          // (b*H + h)*L1 + l1
    const int l1   = prob % L1_;
    const int bh   = prob / L1_;               // b*H + h
    const int b    = bh / H_;

    const float*   Qp = Q + (size_t)prob * L2_ * DK_;
    const float*   Kp = K + (size_t)prob * L2_ * DK_;
    const float*   Vp = V + (size_t)prob * L2_ * DK_;
    const uint8_t* Mp = M + (size_t)(b * L1_ + l1) * L2_ * L2_;
    float*         Op = O + (size_t)prob * L2_ * DK_;

    const int q0 = qt * 16;

    // ---- Q tile in WMMA A layout: chunk kc -> Q[q0+ln][4*kc + 2*g + {0,1}]
    v2f qa[8];
    {
        const float* qrow = Qp + (size_t)(q0 + ln) * DK_ + 2 * g;
        #pragma unroll
        for (int kc = 0; kc < 8; ++kc)
            qa[kc] = *(const v2f*)(qrow + 4 * kc);
    }

    v8f o0 = {}, o1 = {};                 // O accumulators, dk cols [0,16)/[16,32)
    float mrow[8], lrow[8];               // per-row running max/sum (row = r + 8*g)
    #pragma unroll
    for (int r = 0; r < 8; ++r) { mrow[r] = -__builtin_inff(); lrow[r] = 0.0f; }

    for (int kg = 0; kg < NKG_; ++kg) {
        const int k0 = kg * (16 * FT_);

        // ---- S sub-tiles: S[t] = Q * K^T (B operand addressing == A pattern)
        v8f s[FT_] = {};
        #pragma unroll
        for (int t = 0; t < FT_; ++t) {
            const float* krow = Kp + (size_t)(k0 + 16 * t + ln) * DK_ + 2 * g;
            #pragma unroll
            for (int kc = 0; kc < 8; ++kc) {
                v2f kb = *(const v2f*)(krow + 4 * kc);
                s[t] = __builtin_amdgcn_wmma_f32_16x16x4_f32(
                           false, qa[kc], false, kb, (short)0, s[t], false, false);
            }
        }

        // ---- scale (log2 domain) + mask (C layout: row = r+8g, col = ln)
        const uint8_t* mbase = Mp + (size_t)q0 * L2_ + (size_t)(k0 + ln);
        #pragma unroll
        for (int r = 0; r < 8; ++r) {
            const uint8_t* mb = mbase + (size_t)(r + 8 * g) * L2_;
            #pragma unroll
            for (int t = 0; t < FT_; ++t) {
                float sv = s[t][r] * SCALE2_;
                s[t][r] = mb[16 * t] ? NEGINF2_ : sv;
            }
        }

        // ---- online softmax over 64 keys, DPP butterflies in 16-lane halves
        #pragma unroll
        for (int r = 0; r < 8; ++r) {
            float rmax = s[0][r];
            #pragma unroll
            for (int t = 1; t < FT_; ++t) rmax = fmaxf(rmax, s[t][r]);
            rmax = rowmax16(rmax);
            float mnew = fmaxf(mrow[r], rmax);

            float psum = 0.0f;
            #pragma unroll
            for (int t = 0; t < FT_; ++t) {
                float p = exp2f(s[t][r] - mnew);
                s[t][r] = p;
                psum += p;
            }
            psum = rowsum16(psum);

            float corr = exp2f(mrow[r] - mnew);
            lrow[r] = lrow[r] * corr + psum;
            mrow[r] = mnew;
            o0[r] *= corr;
            o1[r] *= corr;
        }

        // ---- P (C layout) -> LDS; per-wave DS ops execute in order, so no
        // barrier is needed (single-wave workgroup).
        #pragma unroll
        for (int t = 0; t < FT_; ++t)
            #pragma unroll
            for (int r = 0; r < 8; ++r)
                Pl[t][r + 8 * g][ln] = s[t][r];

        // ---- O += P * V : per sub-tile, 4 K-chunks x 2 dk blocks
        #pragma unroll
        for (int t = 0; t < FT_; ++t) {
            const int kb0 = k0 + 16 * t;
            #pragma unroll
            for (int kc = 0; kc < 4; ++kc) {
                v2f pa = *(const v2f*)&Pl[t][ln][4 * kc + 2 * g];
                const float* vrow = Vp + (size_t)(kb0 + 4 * kc + 2 * g) * DK_ + ln;
                v2f vb0, vb1;
                vb0.x = vrow[0];        vb0.y = vrow[DK_];
                vb1.x = vrow[16];       vb1.y = vrow[DK_ + 16];
                o0 = __builtin_amdgcn_wmma_f32_16x16x4_f32(
                         false, pa, false, vb0, (short)0, o0, false, false);
                o1 = __builtin_amdgcn_wmma_f32_16x16x4_f32(
                         false, pa, false, vb1, (short)0, o1, false, false);
            }
        }
    }

    // ---- normalize and store: out[q0+r+8g][ln] and [ln+16]
    #pragma unroll
    for (int r = 0; r < 8; ++r) {
        float inv = 1.0f / lrow[r];
        float* orow = Op + (size_t)(q0 + r + 8 * g) * DK_ + ln;
        orow[0]  = o0[r] * inv;
        orow[16] = o1[r] * inv;
    }
}

extern "C" void kernel_launch(void* const* d_in, const int* in_sizes, int n_in,
                              void* d_out, int out_size, void* d_ws, size_t ws_size,
                              hipStream_t stream) {
    (void)in_sizes; (void)n_in; (void)out_size; (void)d_ws; (void)ws_size;
    const float*   Q = (const float*)d_in[0];
    const float*   K = (const float*)d_in[1];
    const float*   V = (const float*)d_in[2];
    const uint8_t* M = (const uint8_t*)d_in[3];
    float*         O = (float*)d_out;

    dim3 grid(B_ * H_ * L1_ * QT_);   // 8192 single-wave workgroups
    dim3 block(32);
    sdpa_f32_wmma_kernel<<<grid, block, 0, stream>>>(Q, K, V, M, O);
}